// CustomLoss_43645457662200
// MI455X (gfx1250) — compile-verified
//
#include <hip/hip_runtime.h>
#include <hip/hip_bf16.h>

typedef float __attribute__((ext_vector_type(2))) v2f;
typedef float __attribute__((ext_vector_type(4))) vf4;
typedef float __attribute__((ext_vector_type(8))) v8f;
typedef int   __attribute__((ext_vector_type(4))) vi4;

// Per-sample fused loss term (coefficients folded in; final scale is 1/B).
// Fully branchless: both relu arms are computed and selected via v_cndmask,
// so the hot loop has no exec-mask manipulation.
__device__ __forceinline__ float loss_term(float l, float u, float t, float pp,
                                           int dtv, int pvv) {
    float c  = (l + u) * 0.5f;
    float d  = t - c;
    float w  = u - l;                       // width (can be negative)
    float vp = fmaxf(-w, 0.0f);             // max(lower-upper, 0)
    float up = fmaxf(c - pp, 0.0f);
    float dn = fmaxf(pp - c, 0.0f);
    float dir = (pvv == 0) ? up : dn;       // v_cndmask
    dir = (dtv != 0) ? dir : 0.0f;          // v_cndmask
    // 1.5*d^2 + 0.1*w + 10*vp + 0.5*dir
    return fmaf(1.5f * d, d, fmaf(0.1f, w, fmaf(10.0f, vp, 0.5f * dir)));
}

__global__ void __launch_bounds__(256)
loss_partial_kernel(const float* __restrict__ pred,
                    const float* __restrict__ target,
                    const float* __restrict__ prev_pci,
                    const int*  __restrict__ delta_time,
                    const int*  __restrict__ pv_values,
                    float* __restrict__ block_partials,
                    long long n) {
    float acc = 0.0f;

    const long long stride = (long long)gridDim.x * blockDim.x * 4;
    for (long long base = ((long long)blockIdx.x * blockDim.x + threadIdx.x) * 4;
         base < n; base += stride) {
        if (base + 3 < n) {
            // Fully vectorized path: b128 non-temporal streaming loads
            // (single-pass data, no reuse -> don't pollute L2).
            vf4 p0 = __builtin_nontemporal_load((const vf4*)(pred + 2 * base));
            vf4 p1 = __builtin_nontemporal_load((const vf4*)(pred + 2 * base + 4));
            vf4 tg = __builtin_nontemporal_load((const vf4*)(target + base));
            vf4 pp = __builtin_nontemporal_load((const vf4*)(prev_pci + base));
            vi4 dt = __builtin_nontemporal_load((const vi4*)(delta_time + base));
            vi4 pv = __builtin_nontemporal_load((const vi4*)(pv_values + base));

            acc += loss_term(p0.x, p0.y, tg.x, pp.x, dt.x, pv.x);
            acc += loss_term(p0.z, p0.w, tg.y, pp.y, dt.y, pv.y);
            acc += loss_term(p1.x, p1.y, tg.z, pp.z, dt.z, pv.z);
            acc += loss_term(p1.z, p1.w, tg.w, pp.w, dt.w, pv.w);
        } else {
            for (long long i = base; i < n; ++i) {
                acc += loss_term(pred[2 * i], pred[2 * i + 1], target[i],
                                 prev_pci[i], delta_time[i], pv_values[i]);
            }
        }
    }

    // ---- Wave32 reduction via V_WMMA_F32_16X16X4_F32 ----
    // A (16x4 f32): lanes 0-15 supply K=0 (vgpr0) / K=1 (vgpr1) of row M=lane,
    //               lanes 16-31 supply K=2 / K=3 of row M=lane-16.
    // With B = all-ones, D[m][n] = sum_k A[m][k] = acc[m] + acc[m+16]
    // regardless of B's lane layout. EXEC is all-ones here (no divergence).
    v2f a; a.x = acc;  a.y = 0.0f;
    v2f b; b.x = 1.0f; b.y = 1.0f;
    v8f c = {};
    c = __builtin_amdgcn_wmma_f32_16x16x4_f32(
            /*neg_a=*/false, a, /*neg_b=*/false, b,
            /*c_mod=*/(short)0, c, /*reuse_a=*/false, /*reuse_b=*/false);

    // Lanes 0-15: D rows 0-7 in vgprs 0-7 -> sum covers acc[0..7]+acc[16..23].
    // Lanes 16-31: D rows 8-15           -> sum covers acc[8..15]+acc[24..31].
    float s = ((c[0] + c[1]) + (c[2] + c[3])) + ((c[4] + c[5]) + (c[6] + c[7]));

    __shared__ float wsum[16];             // 8 waves x 2 half-sums
    const int lane = threadIdx.x & 31;
    const int wave = threadIdx.x >> 5;
    if ((lane & 15) == 0) {
        wsum[(wave << 1) | (lane >> 4)] = s;
    }
    __syncthreads();

    if (threadIdx.x == 0) {
        float bsum = 0.0f;
        #pragma unroll
        for (int i = 0; i < 16; ++i) bsum += wsum[i];
        block_partials[blockIdx.x] = bsum;   // every block writes its own slot
    }
}

__global__ void __launch_bounds__(256)
loss_finalize_kernel(const float* __restrict__ block_partials, int nparts,
                     float* __restrict__ out, float inv_n) {
    __shared__ float sm[256];
    float s = 0.0f;
    for (int i = threadIdx.x; i < nparts; i += 256) s += block_partials[i];
    sm[threadIdx.x] = s;
    __syncthreads();
    for (int off = 128; off > 0; off >>= 1) {
        if ((int)threadIdx.x < off) sm[threadIdx.x] += sm[threadIdx.x + off];
        __syncthreads();
    }
    if (threadIdx.x == 0) out[0] = sm[0] * inv_n;
}

extern "C" void kernel_launch(void* const* d_in, const int* in_sizes, int n_in,
                              void* d_out, int out_size, void* d_ws, size_t ws_size,
                              hipStream_t stream) {
    const float* pred       = (const float*)d_in[0];
    const float* target     = (const float*)d_in[1];
    const float* prev_pci   = (const float*)d_in[2];
    const int*   delta_time = (const int*)d_in[3];
    const int*   pv_values  = (const int*)d_in[4];

    const long long n = (long long)in_sizes[3];   // B (delta_time is (B,))

    const int threads = 256;                      // 8 wave32s per block
    long long want = (n + (long long)(threads * 4) - 1) / (threads * 4);
    int grid = (int)(want < 2048 ? want : 2048);
    if (grid < 1) grid = 1;
    // Clamp to workspace capacity (one float partial per block).
    long long cap = (long long)(ws_size / sizeof(float));
    if (cap >= 1 && grid > cap) grid = (int)cap;

    float* partials = (float*)d_ws;

    loss_partial_kernel<<<grid, threads, 0, stream>>>(
        pred, target, prev_pci, delta_time, pv_values, partials, n);

    loss_finalize_kernel<<<1, 256, 0, stream>>>(
        partials, grid, (float*)d_out, 1.0f / (float)n);
}